// ConvCNPWeather_46437186404663
// MI455X (gfx1250) — compile-verified
//
#include <hip/hip_runtime.h>

// ---------------------------------------------------------------------------
// OffToOn RBF SetConv (ConvCNP encoder) for MI455X / gfx1250.
//   out[b, 0:26,  g, h] = data / max(density, 1e-8)
//   out[b, 26:52, g, h] = density
// Pipeline:
//   pass 1: kx[b][g][n], ky[b][n][h] RBF tables + finite-masked wm[b][52][n]
//           (all stay hot in the 192MB L2)
//   pass 2: one workgroup per (b,g): D[64,128] = (wm .* kx)[64,2048] x Ky[2048,128]
//           on the f32 WMMA pipe (V_WMMA_F32_16X16X4_F32). K is chunked by 64;
//           chunk staging uses GLOBAL_LOAD_ASYNC_TO_LDS_B128 (ASYNCcnt) with
//           double-buffered LDS so the next chunk's fetch overlaps WMMA.
// ---------------------------------------------------------------------------

typedef __attribute__((ext_vector_type(2))) float v2f;
typedef __attribute__((ext_vector_type(8))) float v8f;

#define INV2L2 500000.0f   // 0.5 / (0.001*0.001)

constexpr int Bn = 2;
constexpr int C  = 26;
constexpr int NS = 2048;   // stations (K dimension)
constexpr int G1 = 240;    // lon grid
constexpr int G2 = 121;    // lat grid (N dimension)
constexpr int C2 = 2 * C;  // 52 rows: data + density
constexpr int MP = 64;     // padded M
constexpr int HP = 128;    // padded N
constexpr int KC = 64;     // K chunk staged in LDS
constexpr int NCHUNK = NS / KC;

constexpr int SWM_STRIDE = 68;   // [m][k]; 68%64=4 -> conflict-free A frags; row = 272B (16B aligned)
constexpr int SKY_STRIDE = 144;  // [k][h]; 144%64=16, 2*144%64=32 -> conflict-free B frags; row = 576B
constexpr int SD_STRIDE  = 132;  // epilogue tile spill

constexpr int WM_FLOATS = MP * SWM_STRIDE;  // 4352
constexpr int KY_FLOATS = KC * SKY_STRIDE;  // 9216
constexpr int KY_BASE   = 2 * WM_FLOATS;                  // float offset of sKY[0]
constexpr int KX_BASE   = 2 * WM_FLOATS + 2 * KY_FLOATS;  // float offset of sKX[0]
constexpr int LDS_FLOATS = KX_BASE + 2 * KC;              // 27264 floats = 109056 B

__device__ __forceinline__ bool finitef(float x) {
  union { float f; unsigned u; } v; v.f = x;
  return (v.u & 0x7f800000u) != 0x7f800000u;
}

// 32-bit LDS address of a generic pointer into LDS. The empty asm with "+v"
// keeps this a runtime VGPR value so no addrspacecast constant can leak into
// a static initializer (which ld.lld cannot relocate).
__device__ __forceinline__ unsigned lds_addr_u32(const float* p) {
  unsigned lo = (unsigned)(unsigned long long)p;
  asm volatile("" : "+v"(lo));
  return lo;
}

// Async 16B global -> LDS copy, tracked by ASYNCcnt (gfx1250).
__device__ __forceinline__ void async_b128(unsigned lds_off, const float* gptr) {
  asm volatile("global_load_async_to_lds_b128 %0, %1, off"
               :: "v"(lds_off), "v"((unsigned long long)gptr)
               : "memory");
}
__device__ __forceinline__ void wait_async0() {
  asm volatile("s_wait_asynccnt 0x0" ::: "memory");
}

// kx[b][g][n] = exp(-inv2l2 * (x_lon[b,n] - grid_lon[g])^2)
__global__ void rbf_kx_kernel(const float* __restrict__ x_lon,
                              const float* __restrict__ grid_lon,
                              float* __restrict__ kx) {
  int idx = blockIdx.x * 256 + threadIdx.x;
  if (idx >= Bn * G1 * NS) return;
  int n  = idx & (NS - 1);
  int bg = idx >> 11;
  int g  = bg % G1;
  int b  = bg / G1;
  float d = x_lon[b * NS + n] - grid_lon[g];
  kx[idx] = __expf(-INV2L2 * d * d);
}

// ky[b][n][h] = exp(-inv2l2 * (x_lat[b,n] - grid_lat[h])^2), zero-padded h>=121
__global__ void rbf_ky_kernel(const float* __restrict__ x_lat,
                              const float* __restrict__ grid_lat,
                              float* __restrict__ ky) {
  int idx = blockIdx.x * 256 + threadIdx.x;
  if (idx >= Bn * NS * HP) return;
  int h  = idx & (HP - 1);
  int bn = idx >> 7;
  int n  = bn & (NS - 1);
  int b  = bn >> 11;
  float v = 0.0f;
  if (h < G2) {
    float d = x_lat[b * NS + n] - grid_lat[h];
    v = __expf(-INV2L2 * d * d);
  }
  ky[idx] = v;
}

// wm[b][m][n]: m<26 -> finite? wt : 0 ; m in 26..51 -> finite? 1 : 0
__global__ void wm_kernel(const float* __restrict__ wt,
                          float* __restrict__ wm) {
  int idx = blockIdx.x * 256 + threadIdx.x;
  if (idx >= Bn * C2 * NS) return;
  int n  = idx & (NS - 1);
  int bm = idx >> 11;
  int m  = bm % C2;
  int b  = bm / C2;
  int c  = (m < C) ? m : (m - C);
  float wv = wt[((size_t)b * C + c) * NS + n];
  bool fin = finitef(wv);
  wm[idx] = (m < C) ? (fin ? wv : 0.0f) : (fin ? 1.0f : 0.0f);
}

__launch_bounds__(256)
__global__ void setconv_wmma_kernel(const float* __restrict__ wm,
                                    const float* __restrict__ kx,
                                    const float* __restrict__ ky,
                                    float* __restrict__ out) {
  __shared__ float lds[LDS_FLOATS];

  const int bid = blockIdx.x;
  const int b = bid / G1;
  const int g = bid % G1;

  const int tid   = threadIdx.x;
  const int wave  = tid >> 5;          // wave32
  const int lane  = tid & 31;
  const int mt    = wave >> 1;         // m-tile 0..3
  const int nb    = (wave & 1) * 4;    // n-tile base: 0 or 4
  const int ml    = lane & 15;
  const int klane = (lane >> 4) << 1;  // lanes 0-15: K=k,k+1 ; lanes 16-31: K=k+2,k+3

  const float* kxcol = kx + ((size_t)b * G1 + g) * NS;
  const float* kyb   = ky + (size_t)b * NS * HP;
  const float* wmb   = wm + (size_t)b * C2 * NS;

  // Runtime 32-bit LDS base address (byte units) for async-copy destinations.
  const unsigned ldsU = lds_addr_u32(lds);

  // Rows 52..63 of both sWM buffers are zero-padding; async copies never touch
  // them, so write them exactly once.
  for (int i = tid; i < (MP - C2) * KC; i += 256) {
    int m = C2 + (i >> 6);
    int k = i & (KC - 1);
    lds[0 * WM_FLOATS + m * SWM_STRIDE + k] = 0.0f;
    lds[1 * WM_FLOATS + m * SWM_STRIDE + k] = 0.0f;
  }

  // ---- async chunk stage: wm tile [52x64], ky tile [64x128], kx slice [64] ----
  auto issue_chunk = [&](int k0, int buf) {
    const unsigned wmL = ldsU + (unsigned)(buf * WM_FLOATS) * 4u;
    const unsigned kyL = ldsU + (unsigned)(KY_BASE + buf * KY_FLOATS) * 4u;
    const unsigned kxL = ldsU + (unsigned)(KX_BASE + buf * KC) * 4u;
    // ky: 64 rows x 32 b128 segments
    for (int s = tid; s < KC * 32; s += 256) {
      int row = s >> 5, seg = s & 31;
      async_b128(kyL + (unsigned)(row * SKY_STRIDE + seg * 4) * 4u,
                 kyb + (size_t)(k0 + row) * HP + seg * 4);
    }
    // wm: 52 rows x 16 b128 segments
    for (int s = tid; s < C2 * 16; s += 256) {
      int row = s >> 4, seg = s & 15;
      async_b128(wmL + (unsigned)(row * SWM_STRIDE + seg * 4) * 4u,
                 wmb + (size_t)row * NS + k0 + seg * 4);
    }
    // kx: 16 b128 segments
    if (tid < 16) {
      async_b128(kxL + (unsigned)tid * 16u, kxcol + k0 + tid * 4);
    }
  };

  v8f acc[4] = {};

  issue_chunk(0, 0);
  wait_async0();
  __syncthreads();   // also covers the one-time zero-padding stores

  for (int ci = 0; ci < NCHUNK; ++ci) {
    const int cur = ci & 1;
    if (ci + 1 < NCHUNK) issue_chunk((ci + 1) * KC, cur ^ 1);

    const float* wmRow = lds + cur * WM_FLOATS + (mt * 16 + ml) * SWM_STRIDE;
    const float* kxs   = lds + KX_BASE + cur * KC;
    const float* kyt   = lds + KY_BASE + cur * KY_FLOATS;

    #pragma unroll 4
    for (int k = 0; k < KC; k += 4) {
      int kk = k + klane;
      // A fragment: fold the per-station kx factor in at fragment-load time
      v2f a;
      a.x = wmRow[kk]     * kxs[kk];
      a.y = wmRow[kk + 1] * kxs[kk + 1];
      const float* b0p = kyt + kk * SKY_STRIDE;
      const float* b1p = b0p + SKY_STRIDE;
      #pragma unroll
      for (int t = 0; t < 4; ++t) {
        int col = (nb + t) * 16 + ml;
        v2f bf;
        bf.x = b0p[col];
        bf.y = b1p[col];
        acc[t] = __builtin_amdgcn_wmma_f32_16x16x4_f32(
            false, a, false, bf, (short)0, acc[t], false, false);
      }
    }

    wait_async0();     // next chunk's async copies have landed in LDS
    __syncthreads();   // ...and are visible to all waves; LDS safe to recycle
  }

  // Spill 16x16 f32 accumulator tiles (VGPR j: lanes0-15 -> M=j, lanes16-31 -> M=j+8)
  // sD overlays the staging LDS: [MP][SD_STRIDE] = 8448 floats.
  {
    int rowAdd = (lane < 16) ? 0 : 8;
    int colLo  = lane & 15;
    #pragma unroll
    for (int t = 0; t < 4; ++t) {
      #pragma unroll
      for (int j = 0; j < 8; ++j) {
        int row = mt * 16 + j + rowAdd;
        int col = (nb + t) * 16 + colLo;
        lds[row * SD_STRIDE + col] = acc[t][j];
      }
    }
  }
  __syncthreads();

  // Density-normalize and store: out[b][c2][g][h]
  for (int i = tid; i < C2 * G2; i += 256) {
    int c2 = i / G2;
    int h  = i - c2 * G2;
    float v = lds[c2 * SD_STRIDE + h];
    if (c2 < C) {
      float den = lds[(c2 + C) * SD_STRIDE + h];
      v = v / fmaxf(den, 1e-8f);
    }
    out[(((size_t)b * C2 + c2) * G1 + g) * G2 + h] = v;
  }
}

extern "C" void kernel_launch(void* const* d_in, const int* in_sizes, int n_in,
                              void* d_out, int out_size, void* d_ws, size_t ws_size,
                              hipStream_t stream) {
  (void)in_sizes; (void)n_in; (void)out_size; (void)ws_size;
  const float* x_lon    = (const float*)d_in[0];
  const float* x_lat    = (const float*)d_in[1];
  const float* wt       = (const float*)d_in[2];
  const float* grid_lon = (const float*)d_in[3];
  const float* grid_lat = (const float*)d_in[4];
  float* out = (float*)d_out;

  float* kx_ws = (float*)d_ws;                        // [B][G1][NS]  ~3.9 MB
  float* ky_ws = kx_ws + (size_t)Bn * G1 * NS;        // [B][NS][HP]  ~2.0 MB
  float* wm_ws = ky_ws + (size_t)Bn * NS * HP;        // [B][52][NS]  ~0.85 MB

  rbf_kx_kernel<<<(Bn * G1 * NS + 255) / 256, 256, 0, stream>>>(x_lon, grid_lon, kx_ws);
  rbf_ky_kernel<<<(Bn * NS * HP + 255) / 256, 256, 0, stream>>>(x_lat, grid_lat, ky_ws);
  wm_kernel<<<(Bn * C2 * NS + 255) / 256, 256, 0, stream>>>(wt, wm_ws);
  setconv_wmma_kernel<<<Bn * G1, 256, 0, stream>>>(wm_ws, kx_ws, ky_ws, out);
}